// PortfolioModel_30580167147948
// MI455X (gfx1250) — compile-verified
//
#include <hip/hip_runtime.h>

// ---------------------------------------------------------------------------
// Portfolio model on MI455X (gfx1250, wave32).
// Each wave handles a 16-row batch tile. The three 5x5 matmuls per layer are
// computed transposed (T' = W * T, T = 5x16 tile) with V_WMMA_F32_16X16X4_F32,
// K=5 split as two K=4 WMMAs (K 0..3 and 4..7, padded with zeros).
// D-layout (row M = VGPR, col N = lane) leaves each lane holding the 5 asset
// values of one batch row -> softmax / rebalance / wealth run per-lane.
// This revision removes divergent predicated loads (clamped-index loads +
// cndmask) and IEEE division expansion (v_rcp_f32 instead).
// ---------------------------------------------------------------------------

#define PDIM 5

static constexpr float kLB = 0.0f;
static constexpr float kUB = 0.5f;
static constexpr float kRF = 1.00075f;
static constexpr int   kNIters = 8;

typedef float v2f __attribute__((ext_vector_type(2)));
typedef float v8f __attribute__((ext_vector_type(8)));

__device__ __forceinline__ float fast_rcp(float x) {
    return __builtin_amdgcn_rcpf(x);          // v_rcp_f32 (~1 ulp)
}

__device__ __forceinline__ float tanh_fast(float x) {
    // tanh via hardware v_exp_f32 + v_rcp_f32; clamp avoids inf*0.
    float xc = fminf(fmaxf(x, -10.0f), 10.0f);
    float e  = __expf(2.0f * xc);
    return (e - 1.0f) * fast_rcp(e + 1.0f);
}

__device__ __forceinline__ float clamp01ub(float v) {
    return fminf(fmaxf(v, kLB), kUB);
}

// One portfolio layer for a 16-row tile. `row` = this lane's batch row
// (lanes >=16 mirror lanes 0-15). Returns updated wealth for this lane's row.
__device__ __forceinline__ float layer_eval(
    float xw,
    const float* __restrict__ R,
    const float* __restrict__ W1,
    const float* __restrict__ W2,
    const float* __restrict__ W3,
    int row, int lane)
{
    const bool hi = (lane >= 16);
    const int  m  = lane & 15;
    const int  mc = (m < PDIM) ? m : (PDIM - 1);   // clamped row index into W
    const bool valid = (m < PDIM);

    // ---- Load R tile directly in WMMA B-form (5x16, K split over halves) ----
    const float* Rr = R + (long long)row * PDIM;
    float rb0 = Rr[hi ? 2 : 0];             // B[K=0|2][n]
    float rb1 = Rr[hi ? 3 : 1];             // B[K=1|3][n]
    float r4l = Rr[4];                      // unconditional, in-bounds
    float rb2 = hi ? 0.0f : r4l;            // B[K=4|6][n] (K=6 row is zero)

    // Full per-lane R row for the wealth update (mirror into high half).
    float r0 = __shfl(rb0, m, 32);
    float r1 = __shfl(rb1, m, 32);
    float r2 = __shfl(rb0, m + 16, 32);
    float r3 = __shfl(rb1, m + 16, 32);
    float r4 = __shfl(rb2, m, 32);

    v2f b1; b1.x = rb0; b1.y = rb1;
    v2f b2; b2.x = rb2; b2.y = 0.0f;

    float h0 = 0.f, h1 = 0.f, h2 = 0.f, h3 = 0.f, h4 = 0.f;
    const float* Ws[3] = { W1, W2, W3 };

#pragma unroll
    for (int s = 0; s < 3; ++s) {
        const float* W = Ws[s];
        // Unconditional in-bounds loads (clamped row), then mask with select.
        float w01 = W[mc * PDIM + (hi ? 2 : 0)];
        float w23 = W[mc * PDIM + (hi ? 3 : 1)];
        float w4  = W[mc * PDIM + 4];
        // A-form of padded 16x8 W: lane%16 = M, lane-half selects K-block.
        v2f a1, a2;
        a1.x = valid ? w01 : 0.0f;
        a1.y = valid ? w23 : 0.0f;
        a2.x = (valid && !hi) ? w4 : 0.0f;                // K=4 (K=6 -> 0)
        a2.y = 0.0f;                                      // K=5 / K=7 -> 0

        v8f c = {};
        // D = Wpad(16x8) x Tpad(8x16): two K=4 WMMAs, fp32 exact.
        c = __builtin_amdgcn_wmma_f32_16x16x4_f32(false, a1, false, b1,
                                                  (short)0, c, false, false);
        c = __builtin_amdgcn_wmma_f32_16x16x4_f32(false, a2, false, b2,
                                                  (short)0, c, false, false);

        // tanh on the 5 live rows (rows 5..15 are exactly zero; tanh(0)=0).
        h0 = tanh_fast(c[0]);
        h1 = tanh_fast(c[1]);
        h2 = tanh_fast(c[2]);
        h3 = tanh_fast(c[3]);
        h4 = tanh_fast(c[4]);

        // Rebuild B-form for next stage: rows 2,3 move to the high lane-half.
        float s2 = __shfl(h2, m, 32);
        float s3 = __shfl(h3, m, 32);
        b1.x = hi ? s2 : h0;
        b1.y = hi ? s3 : h1;
        b2.x = hi ? 0.0f : h4;
        b2.y = 0.0f;
    }

    // ---- softmax over the 5 assets (per lane) ----
    float mx = fmaxf(fmaxf(fmaxf(h0, h1), fmaxf(h2, h3)), h4);
    float e0 = __expf(h0 - mx), e1 = __expf(h1 - mx), e2 = __expf(h2 - mx);
    float e3 = __expf(h3 - mx), e4 = __expf(h4 - mx);
    float inv = fast_rcp(e0 + e1 + e2 + e3 + e4);
    float w0 = e0 * inv, w1 = e1 * inv, w2 = e2 * inv, w3 = e3 * inv, w4 = e4 * inv;

    // ---- rebalance: clamp + proportional redistribution (8 fixed iters) ----
    float o0 = w0, o1 = w1, o2 = w2, o3 = w3, o4 = w4;
    float n0 = clamp01ub(w0), n1 = clamp01ub(w1), n2 = clamp01ub(w2);
    float n3 = clamp01ub(w3), n4 = clamp01ub(w4);
#pragma unroll
    for (int it = 0; it < kNIters; ++it) {
        float leftover = (o0 - n0) + (o1 - n1) + (o2 - n2) + (o3 - n3) + (o4 - n4);
        bool m0 = (n0 != kUB), m1 = (n1 != kUB), m2 = (n2 != kUB);
        bool m3 = (n3 != kUB), m4 = (n4 != kUB);
        float osum = (m0 ? n0 : 0.f) + (m1 ? n1 : 0.f) + (m2 ? n2 : 0.f)
                   + (m3 ? n3 : 0.f) + (m4 ? n4 : 0.f);
        float invs = leftover * fast_rcp(osum);
        n0 += m0 ? invs * n0 : 0.f;
        n1 += m1 ? invs * n1 : 0.f;
        n2 += m2 ? invs * n2 : 0.f;
        n3 += m3 ? invs * n3 : 0.f;
        n4 += m4 ? invs * n4 : 0.f;
        o0 = n0; o1 = n1; o2 = n2; o3 = n3; o4 = n4;
        bool exceed = (n0 > kUB) || (n1 > kUB) || (n2 > kUB) || (n3 > kUB) || (n4 > kUB);
        n0 = exceed ? clamp01ub(n0) : n0;
        n1 = exceed ? clamp01ub(n1) : n1;
        n2 = exceed ? clamp01ub(n2) : n2;
        n3 = exceed ? clamp01ub(n3) : n3;
        n4 = exceed ? clamp01ub(n4) : n4;
    }

    // ---- wealth update: x * Rf * sum_p w_p * (1 + R_p) ----
    float acc = n0 * (1.0f + r0) + n1 * (1.0f + r1) + n2 * (1.0f + r2)
              + n3 * (1.0f + r3) + n4 * (1.0f + r4);
    return xw * acc * kRF;
}

__global__ void __launch_bounds__(128)
portfolio_kernel(const float* __restrict__ x,
                 const float* __restrict__ R1,
                 const float* __restrict__ R2,
                 const float* __restrict__ W11, const float* __restrict__ W12,
                 const float* __restrict__ W13,
                 const float* __restrict__ W21, const float* __restrict__ W22,
                 const float* __restrict__ W23,
                 float* __restrict__ partials, int batch)
{
    const int lane = threadIdx.x & 31;
    const int wave = threadIdx.x >> 5;                 // 0..3
    const int tile = blockIdx.x * 4 + wave;            // 16 rows per tile
    const int m    = lane & 15;
    const int row0 = tile * 16 + m;
    const bool live = (lane < 16) && (row0 < batch);
    const int row  = min(row0, batch - 1);             // keep loads in-bounds, EXEC full

    float xw = x[row];
    float x1 = layer_eval(xw, R1, W11, W12, W13, row, lane);
    float x2 = layer_eval(x1, R2, W21, W22, W23, row, lane);

    // deterministic reduction: wave -> block -> partials[blockIdx]
    float contrib = live ? x2 : 0.0f;
#pragma unroll
    for (int off = 16; off > 0; off >>= 1)
        contrib += __shfl_down(contrib, off, 32);

    __shared__ float sm[4];
    if (lane == 0) sm[wave] = contrib;
    __syncthreads();
    if (threadIdx.x == 0)
        partials[blockIdx.x] = (sm[0] + sm[1]) + (sm[2] + sm[3]);
}

__global__ void __launch_bounds__(256)
reduce_kernel(const float* __restrict__ partials, int n,
              float* __restrict__ out, float scale)
{
    __shared__ float sm[256];
    float s = 0.0f;
    for (int i = threadIdx.x; i < n; i += 256) s += partials[i];
    sm[threadIdx.x] = s;
    __syncthreads();
#pragma unroll
    for (int off = 128; off > 0; off >>= 1) {
        if ((int)threadIdx.x < off) sm[threadIdx.x] += sm[threadIdx.x + off];
        __syncthreads();
    }
    if (threadIdx.x == 0) out[0] = sm[0] * scale;
}

extern "C" void kernel_launch(void* const* d_in, const int* in_sizes, int n_in,
                              void* d_out, int out_size, void* d_ws, size_t ws_size,
                              hipStream_t stream) {
    const float* x   = (const float*)d_in[0];
    const float* R1  = (const float*)d_in[1];
    const float* R2  = (const float*)d_in[2];
    const float* W11 = (const float*)d_in[3];
    const float* W12 = (const float*)d_in[4];
    const float* W13 = (const float*)d_in[5];
    const float* W21 = (const float*)d_in[6];
    const float* W22 = (const float*)d_in[7];
    const float* W23 = (const float*)d_in[8];

    const int batch  = in_sizes[0];                    // 1,000,000
    const int tiles  = (batch + 15) / 16;              // 62,500
    const int blocks = (tiles + 3) / 4;                // 15,625 (64 rows/block)

    float* partials = (float*)d_ws;                    // blocks * 4 bytes

    portfolio_kernel<<<blocks, 128, 0, stream>>>(
        x, R1, R2, W11, W12, W13, W21, W22, W23, partials, batch);
    reduce_kernel<<<1, 256, 0, stream>>>(partials, blocks, (float*)d_out,
                                         1.0f / (float)batch);
}